// PrototypeLayer_24026047053930
// MI455X (gfx1250) — compile-verified
//
#include <hip/hip_runtime.h>

typedef float v2f __attribute__((ext_vector_type(2)));
typedef float v8f __attribute__((ext_vector_type(8)));

constexpr int Bn = 64;    // batch
constexpr int Cn = 256;   // channels (K)
constexpr int Ln = 1024;  // length  (N)
constexpr int Pn = 512;   // prototypes (M)

// Block tile: 64 (M/P) x 256 (N/L), K chunked by 16, LDS double-buffered.
// LDS strides (floats), padded for bank-conflict-free WMMA fragment reads.
constexpr int ASs = 20;     // As: 64 (M) x 16 (K), row stride 20
constexpr int BSs = 264;    // Bs: 16 (K) x 256 (N), row stride 264

constexpr int AS_FLOATS  = 64 * ASs;             // 1280
constexpr int BS_FLOATS  = 16 * BSs;             // 4224
constexpr int BUF_FLOATS = AS_FLOATS + BS_FLOATS;         // 5504 floats
constexpr unsigned BUF_BYTES     = BUF_FLOATS * 4;        // 22016
constexpr unsigned BS_OFF_BYTES  = AS_FLOATS * 4;         // 5120
constexpr unsigned LDS_TOTAL_BYTES = 2 * BUF_BYTES;       // 44032

// ---------------- p2[p] = sum_c proto[p][c]^2 ----------------
__global__ __launch_bounds__(512) void p2_kernel(const float* __restrict__ proto,
                                                 float* __restrict__ p2) {
    int p = threadIdx.x;             // one block of 512 threads
    const float* row = proto + (size_t)p * Cn;
    float s = 0.f;
    #pragma unroll 8
    for (int c = 0; c < Cn; ++c) { float v = row[c]; s += v * v; }
    p2[p] = s;
}

// ---------------- x2[b][l] = sum_c x[b][c][l]^2 ----------------
__global__ __launch_bounds__(256) void x2_kernel(const float* __restrict__ x,
                                                 float* __restrict__ x2) {
    int idx = blockIdx.x * 256 + threadIdx.x;   // b*Ln + l
    int b = idx >> 10;
    int l = idx & (Ln - 1);
    const float* col = x + (size_t)b * Cn * Ln + l;
    float s = 0.f;
    #pragma unroll 8
    for (int c = 0; c < Cn; ++c) { float v = col[(size_t)c * Ln]; s += v * v; }
    x2[idx] = s;
}

// ---- CDNA5 async global->LDS copy (ASYNCcnt), one 16B packet per lane ----
__device__ __forceinline__ void async_copy_b128(unsigned lds_byte_off, const void* gptr) {
    asm volatile("global_load_async_to_lds_b128 %0, %1, off"
                 :: "v"(lds_byte_off), "v"((unsigned long long)(uintptr_t)gptr)
                 : "memory");
}
__device__ __forceinline__ void wait_async_le5() {
    asm volatile("s_wait_asynccnt 0x5" ::: "memory");
}
__device__ __forceinline__ void wait_async_0() {
    asm volatile("s_wait_asynccnt 0x0" ::: "memory");
}

// ---------------- main fused distance GEMM ----------------
// grid (Ln/256, Pn/64, Bn), block 256 (8 wave32)
// wave tile: 64 (M) x 32 (N) = 4x2 WMMA 16x16 accumulators; each wave owns one N strip.
__global__ __launch_bounds__(256) void proto_dist_kernel(
    const float* __restrict__ x, const float* __restrict__ proto,
    const float* __restrict__ p2, const float* __restrict__ x2,
    float* __restrict__ out) {
    extern __shared__ __align__(16) float smem[];   // 2 * (As + Bs)

    const int b  = blockIdx.z;
    const int p0 = blockIdx.y * 64;
    const int l0 = blockIdx.x * 256;

    const int tid  = threadIdx.x;
    const int wave = tid >> 5;
    const int lane = tid & 31;
    const int ln   = lane & 15;       // N (or M) index within fragment
    const int lhi  = lane >> 4;       // half-wave select

    const int nBase = wave * 32;      // each wave: one 32-wide N strip, all 64 M rows

    v8f acc[4][2];
    #pragma unroll
    for (int i = 0; i < 4; ++i)
        #pragma unroll
        for (int j = 0; j < 2; ++j) acc[i][j] = (v8f){};

    const float* xb = x + (size_t)b * Cn * Ln;

    // per-thread staging coordinates (5 x b128 per thread per chunk)
    // As tile 64x16: 1 float4; Bs tile 16x256: 4 float4s
    const int am  = tid >> 2;            // 0..63
    const int ak  = (tid & 3) << 2;      // 0,4,8,12
    int bk[4], bn[4];
    #pragma unroll
    for (int j = 0; j < 4; ++j) {
        int f = tid + j * 256;  bk[j] = f >> 6;  bn[j] = (f & 63) << 2;
    }

    auto issue_chunk = [&](int c0, int bufIdx) {
        const unsigned bufB = (unsigned)bufIdx * BUF_BYTES;
        async_copy_b128(bufB + (unsigned)((am * ASs + ak) * 4),
                        proto + (size_t)(p0 + am) * Cn + c0 + ak);
        #pragma unroll
        for (int j = 0; j < 4; ++j) {
            async_copy_b128(bufB + BS_OFF_BYTES + (unsigned)((bk[j] * BSs + bn[j]) * 4),
                            xb + (size_t)(c0 + bk[j]) * Ln + l0 + bn[j]);
        }
    };

    issue_chunk(0, 0);

    constexpr int NCHUNK = Cn / 16;     // 16
    #pragma unroll 2
    for (int i = 0; i < NCHUNK; ++i) {
        if (i < NCHUNK - 1) {
            issue_chunk((i + 1) * 16, (i + 1) & 1);   // prefetch next chunk
            wait_async_le5();                         // my 5 loads for chunk i are done
        } else {
            wait_async_0();
        }
        __syncthreads();                              // everyone's chunk-i data visible

        const float* Asb = smem + (i & 1) * BUF_FLOATS;
        const float* Bsb = Asb + AS_FLOATS;

        // ---- 4 K-steps of 4, 8 WMMAs each ----
        #pragma unroll
        for (int kk = 0; kk < 16; kk += 4) {
            // A fragments (16x4 f32): lanes 0-15 hold K=kk,kk+1; lanes 16-31 K=kk+2,kk+3
            v2f a[4];
            #pragma unroll
            for (int mi = 0; mi < 4; ++mi)
                a[mi] = *(const v2f*)&Asb[(mi * 16 + ln) * ASs + kk + lhi * 2];
            // B fragments (4x16 f32): vgpr0 = row kk(+2 for hi half), vgpr1 = next row
            const int krow = kk + lhi * 2;
            v2f b0, b1;
            b0.x = Bsb[(krow    ) * BSs + nBase +      ln];
            b0.y = Bsb[(krow + 1) * BSs + nBase +      ln];
            b1.x = Bsb[(krow    ) * BSs + nBase + 16 + ln];
            b1.y = Bsb[(krow + 1) * BSs + nBase + 16 + ln];

            #pragma unroll
            for (int mi = 0; mi < 4; ++mi) {
                acc[mi][0] = __builtin_amdgcn_wmma_f32_16x16x4_f32(false, a[mi], false, b0,
                               (short)0, acc[mi][0], false, false);
                acc[mi][1] = __builtin_amdgcn_wmma_f32_16x16x4_f32(false, a[mi], false, b1,
                               (short)0, acc[mi][1], false, false);
            }
        }
        __syncthreads();   // all waves done reading buf i before it is refilled
    }

    // ---- epilogue: relu(x2 - 2*xp + p2), C/D layout: vgpr r -> M = r (+8 for hi half) ----
    const float* x2b  = x2 + (size_t)b * Ln;
    float*       outb = out + (size_t)b * Pn * Ln;

    #pragma unroll
    for (int mi = 0; mi < 4; ++mi) {
        const int mg = p0 + mi * 16 + lhi * 8;   // global proto row for vgpr 0
        float p2v[8];
        #pragma unroll
        for (int r = 0; r < 8; ++r) p2v[r] = p2[mg + r];
        #pragma unroll
        for (int ni = 0; ni < 2; ++ni) {
            const int n  = l0 + nBase + ni * 16 + ln;
            const float xv = x2b[n];
            #pragma unroll
            for (int r = 0; r < 8; ++r) {
                float d = xv - 2.0f * acc[mi][ni][r] + p2v[r];
                d = d > 0.0f ? d : 0.0f;
                outb[(size_t)(mg + r) * Ln + n] = d;
            }
        }
    }
}

extern "C" void kernel_launch(void* const* d_in, const int* in_sizes, int n_in,
                              void* d_out, int out_size, void* d_ws, size_t ws_size,
                              hipStream_t stream) {
    const float* x     = (const float*)d_in[0];   // [B, C, L]
    const float* proto = (const float*)d_in[1];   // [P, C]
    float* out = (float*)d_out;                   // [B, P, L]

    float* p2 = (float*)d_ws;                     // Pn floats
    float* x2 = p2 + Pn;                          // Bn*Ln floats

    p2_kernel<<<dim3(1), dim3(512), 0, stream>>>(proto, p2);
    x2_kernel<<<dim3((Bn * Ln) / 256), dim3(256), 0, stream>>>(x, x2);
    proto_dist_kernel<<<dim3(Ln / 256, Pn / 64, Bn), dim3(256), LDS_TOTAL_BYTES, stream>>>(
        x, proto, p2, x2, out);
}